// GCN_74148315398326
// MI455X (gfx1250) — compile-verified
//
#include <hip/hip_runtime.h>
#include <hip/hip_bf16.h>

typedef __attribute__((ext_vector_type(2))) float v2f;
typedef __attribute__((ext_vector_type(8))) float v8f;

#define NREAL 16
#define NEMB  16

// ---------------------------------------------------------------------------
// column-wise max(|rf|) over N rows, 16 columns. abs values are >= 0 so the
// IEEE bit pattern orders like the float; atomicMax on uint is correct.
// ---------------------------------------------------------------------------
__global__ void init_colmax_kernel(unsigned* cmax) {
  if (threadIdx.x < 16) cmax[threadIdx.x] = 0u;
}

__global__ void colmax_abs_kernel(const float* __restrict__ rf,
                                  unsigned* __restrict__ cmax, int N) {
  int tid = blockIdx.x * blockDim.x + threadIdx.x;
  int col = tid & 15;
  int row0 = tid >> 4;
  int stride = (gridDim.x * blockDim.x) >> 4;
  float m = 0.0f;
  for (int r = row0; r < N; r += stride)
    m = fmaxf(m, fabsf(rf[(size_t)r * NREAL + col]));
  atomicMax(&cmax[col], __float_as_uint(m));
}

// ---------------------------------------------------------------------------
// x[i, 0:16]  = rf[i, :] / (colmax + 1e-12)
// x[i, 16:32] = emb_table[cat[i], :]
// ---------------------------------------------------------------------------
__global__ void build_x_kernel(const float* __restrict__ rf,
                               const int* __restrict__ cat,
                               const float* __restrict__ emb,
                               const unsigned* __restrict__ cmax,
                               float* __restrict__ x, int N) {
  int tid = blockIdx.x * blockDim.x + threadIdx.x;
  if (tid >= N * 32) return;
  int i = tid >> 5;
  int d = tid & 31;
  float v;
  if (d < 16) {
    v = rf[(size_t)i * NREAL + d] / (__uint_as_float(cmax[d]) + 1e-12f);
  } else {
    v = emb[(size_t)cat[i] * NEMB + (d - 16)];
  }
  x[tid] = v;
}

// ---------------------------------------------------------------------------
// out(N,O) = relu(A(N,K) @ W(K,O) + bias)  [+ resid, added AFTER the relu]
//
// One wave computes a 64x16 output strip = 4 stacked 16x16 WMMA tiles with
// V_WMMA_F32_16X16X4_F32 (exact fp32 math, matching the fp32 reference).
// K-outer loop: each B (weight) fragment load is reused by 4 WMMAs.
//
// f32 16x16x4 register layout (ISA 7.12.2):
//   A: lane l -> row m=l%16 ; vgpr0/1 = K={0,1} (lanes 0-15) or K={2,3} (16-31)
//   B: lane l -> col n=l%16 ; vgpr0/1 = K rows, same half-split (col-major B)
//   C/D: lane l -> col n=l%16 ; vgpr j -> row m=j (lanes 0-15) / m=8+j (16-31)
// Row indices are clamped on loads and stores are guarded, so EXEC stays
// all-ones through every WMMA (ISA 7.12 requirement).
// ---------------------------------------------------------------------------
__global__ void gemm16_bias_relu_kernel(const float* __restrict__ A,
                                        const float* __restrict__ W,
                                        const float* __restrict__ bias,
                                        const float* __restrict__ resid,
                                        float* __restrict__ out,
                                        int N, int K, int O) {
  const int lane = threadIdx.x;            // 0..31
  const int m0   = blockIdx.x * 64;        // 4 m-tiles of 16 rows
  const int n0   = blockIdx.y * 16;
  const int half = lane >> 4;              // 0 or 1
  const int mn   = lane & 15;

  const float* ap0;
  const float* ap1;
  const float* ap2;
  const float* ap3;
  {
    int r0 = m0 + 0 * 16 + mn; if (r0 > N - 1) r0 = N - 1;
    int r1 = m0 + 1 * 16 + mn; if (r1 > N - 1) r1 = N - 1;
    int r2 = m0 + 2 * 16 + mn; if (r2 > N - 1) r2 = N - 1;
    int r3 = m0 + 3 * 16 + mn; if (r3 > N - 1) r3 = N - 1;
    ap0 = A + (size_t)r0 * K + 2 * half;
    ap1 = A + (size_t)r1 * K + 2 * half;
    ap2 = A + (size_t)r2 * K + 2 * half;
    ap3 = A + (size_t)r3 * K + 2 * half;
  }
  const float* bcol = W + (size_t)(2 * half) * O + (n0 + mn);

  v8f acc0 = {}, acc1 = {}, acc2 = {}, acc3 = {};
  for (int k = 0; k < K; k += 4) {
    v2f b;
    b[0] = bcol[0];
    b[1] = bcol[O];
    v2f a0; a0[0] = ap0[0]; a0[1] = ap0[1];
    v2f a1; a1[0] = ap1[0]; a1[1] = ap1[1];
    v2f a2; a2[0] = ap2[0]; a2[1] = ap2[1];
    v2f a3; a3[0] = ap3[0]; a3[1] = ap3[1];
    acc0 = __builtin_amdgcn_wmma_f32_16x16x4_f32(false, a0, false, b, (short)0, acc0, false, false);
    acc1 = __builtin_amdgcn_wmma_f32_16x16x4_f32(false, a1, false, b, (short)0, acc1, false, false);
    acc2 = __builtin_amdgcn_wmma_f32_16x16x4_f32(false, a2, false, b, (short)0, acc2, false, false);
    acc3 = __builtin_amdgcn_wmma_f32_16x16x4_f32(false, a3, false, b, (short)0, acc3, false, false);
    ap0 += 4; ap1 += 4; ap2 += 4; ap3 += 4;
    bcol += (size_t)4 * O;
  }

  const int n = n0 + mn;
  const float bn = bias[n];

#pragma unroll
  for (int t = 0; t < 4; ++t) {
    const v8f acc = (t == 0) ? acc0 : (t == 1) ? acc1 : (t == 2) ? acc2 : acc3;
#pragma unroll
    for (int j = 0; j < 8; ++j) {
      int m = m0 + t * 16 + half * 8 + j;
      if (m < N) {
        float v = fmaxf(acc[j] + bn, 0.0f);
        if (resid) v += resid[(size_t)m * O + n];
        out[(size_t)m * O + n] = v;
      }
    }
  }
}

// ---------------------------------------------------------------------------
// zero fill
// ---------------------------------------------------------------------------
__global__ void zero_kernel(float* __restrict__ p, long long n) {
  long long tid = (long long)blockIdx.x * blockDim.x + threadIdx.x;
  if (tid < n) p[tid] = 0.0f;
}

// ---------------------------------------------------------------------------
// segment-sum over edges: f[dst[e], :] += msg[src[e], :]
// One thread per (edge, 4-channel chunk): float4 gather + 4 atomic f32 adds.
// Chunk count is a power of two -> shift/mask, no 64-bit division.
// Channel chunk is the fastest-varying index -> coalesced msg reads.
// ---------------------------------------------------------------------------
__global__ void scatter_add_kernel(const float* __restrict__ msg,
                                   const int* __restrict__ src,
                                   const int* __restrict__ dst,
                                   float* __restrict__ f,
                                   long long E, int shift /* O4 = 1<<shift */) {
  long long tid = (long long)blockIdx.x * blockDim.x + threadIdx.x;
  if (tid >= (E << shift)) return;
  long long e = tid >> shift;
  int c4 = (int)(tid & ((1 << shift) - 1));
  int s = src[e];
  int d = dst[e];
  const float4 v = ((const float4*)(msg + ((size_t)s << (shift + 2))))[c4];
  float* fp = f + ((size_t)d << (shift + 2)) + ((size_t)c4 << 2);
  atomicAdd(fp + 0, v.x);
  atomicAdd(fp + 1, v.y);
  atomicAdd(fp + 2, v.z);
  atomicAdd(fp + 3, v.w);
}

// ---------------------------------------------------------------------------
extern "C" void kernel_launch(void* const* d_in, const int* in_sizes, int n_in,
                              void* d_out, int out_size, void* d_ws, size_t ws_size,
                              hipStream_t stream) {
  const float* rf    = (const float*)d_in[0];
  const int*   cat   = (const int*)  d_in[1];
  const int*   src   = (const int*)  d_in[2];
  const int*   dst   = (const int*)  d_in[3];
  const float* emb   = (const float*)d_in[4];
  const float* l1_hid_w  = (const float*)d_in[5];
  const float* l1_hid_b  = (const float*)d_in[6];
  const float* l1_msg_w1 = (const float*)d_in[7];
  const float* l1_msg_b1 = (const float*)d_in[8];
  const float* l1_msg_w2 = (const float*)d_in[9];
  const float* l1_msg_b2 = (const float*)d_in[10];
  const float* l1_agg_w1 = (const float*)d_in[11];
  const float* l1_agg_b1 = (const float*)d_in[12];
  const float* l1_agg_w2 = (const float*)d_in[13];
  const float* l1_agg_b2 = (const float*)d_in[14];
  const float* l2_hid_w  = (const float*)d_in[15];
  const float* l2_hid_b  = (const float*)d_in[16];
  const float* l2_msg_w1 = (const float*)d_in[17];
  const float* l2_msg_b1 = (const float*)d_in[18];
  const float* l2_msg_w2 = (const float*)d_in[19];
  const float* l2_msg_b2 = (const float*)d_in[20];
  const float* l2_agg_w1 = (const float*)d_in[21];
  const float* l2_agg_b1 = (const float*)d_in[22];
  const float* l2_agg_w2 = (const float*)d_in[23];
  const float* l2_agg_b2 = (const float*)d_in[24];

  const int N = in_sizes[0] / NREAL;   // 50000
  const long long E = in_sizes[2];     // 1600000

  // workspace layout (floats)
  float* ws = (float*)d_ws;
  unsigned* cmax = (unsigned*)ws;                        // 16
  float* x    = ws + 16;                                 // N*32
  float* hid1 = x    + (size_t)N * 32;                   // N*128
  float* t64  = hid1 + (size_t)N * 128;                  // N*64
  float* msg1 = t64  + (size_t)N * 64;                   // N*128
  float* f1   = msg1 + (size_t)N * 128;                  // N*128
  float* h1   = f1   + (size_t)N * 128;                  // N*128
  // layer-2 reuses layer-1 regions (all free once h1 is computed)
  float* hid2 = hid1;   // N*32
  float* t64b = t64;    // N*64
  float* msg2 = msg1;   // N*32
  float* f2   = f1;     // N*32
  float* outp = (float*)d_out;  // N*32

  auto gemm = [&](const float* A, const float* W, const float* b,
                  const float* resid, float* o, int K, int O) {
    dim3 grid((N + 63) / 64, O / 16);
    gemm16_bias_relu_kernel<<<grid, 32, 0, stream>>>(A, W, b, resid, o, N, K, O);
  };

  // ---- input preprocessing ----
  init_colmax_kernel<<<1, 16, 0, stream>>>(cmax);
  colmax_abs_kernel<<<128, 256, 0, stream>>>(rf, cmax, N);
  build_x_kernel<<<(N * 32 + 255) / 256, 256, 0, stream>>>(rf, cat, emb, cmax, x, N);

  // ---- GCN layer 1 ----
  gemm(x,    l1_hid_w, l1_hid_b, nullptr, hid1, 32, 128);       // hid1 = relu(x @ W + b)
  gemm(hid1, l1_msg_w1, l1_msg_b1, nullptr, t64, 128, 64);      // msg mlp
  gemm(t64,  l1_msg_w2, l1_msg_b2, nullptr, msg1, 64, 128);
  zero_kernel<<<(unsigned)(((long long)N * 128 + 255) / 256), 256, 0, stream>>>(f1, (long long)N * 128);
  {
    long long total = E << 5;                                   // E * 32 chunks
    scatter_add_kernel<<<(unsigned)((total + 255) / 256), 256, 0, stream>>>(msg1, src, dst, f1, E, 5);
  }
  gemm(f1,  l1_agg_w1, l1_agg_b1, nullptr, t64, 128, 64);       // agg mlp
  gemm(t64, l1_agg_w2, l1_agg_b2, hid1, h1, 64, 128);           // + hid1 after relu

  // ---- GCN layer 2 ----
  gemm(h1,   l2_hid_w, l2_hid_b, nullptr, hid2, 128, 32);       // hid2 = relu(h1 @ W + b)
  gemm(hid2, l2_msg_w1, l2_msg_b1, nullptr, t64b, 32, 64);
  gemm(t64b, l2_msg_w2, l2_msg_b2, nullptr, msg2, 64, 32);
  zero_kernel<<<(unsigned)(((long long)N * 32 + 255) / 256), 256, 0, stream>>>(f2, (long long)N * 32);
  {
    long long total = E << 3;                                   // E * 8 chunks
    scatter_add_kernel<<<(unsigned)((total + 255) / 256), 256, 0, stream>>>(msg2, src, dst, f2, E, 3);
  }
  gemm(f2,   l2_agg_w1, l2_agg_b1, nullptr, t64b, 32, 64);
  gemm(t64b, l2_agg_w2, l2_agg_b2, hid2, outp, 64, 32);         // final: + hid2 after relu
}